// Model_37005438222363
// MI455X (gfx1250) — compile-verified
//
#include <hip/hip_runtime.h>
#include <math.h>

#define KTOP 4
#define HM   64
#define TT   14
#define CCH  128

typedef __attribute__((ext_vector_type(2))) float v2f;
typedef __attribute__((ext_vector_type(8))) float v8f;

// ---------------- Stage 1: per-image top-4 + box construction ----------------
__global__ void topk_boxes_kernel(const float* __restrict__ heat,
                                  const float* __restrict__ off,
                                  const float* __restrict__ bsz,
                                  float* __restrict__ boxesH,
                                  float* __restrict__ boxesV)
{
    const int n   = blockIdx.x;
    const int tid = threadIdx.x;
    const float* h = heat + (size_t)n * (HM * HM);

    __shared__ float sval[256];
    __shared__ int   sidx[256];
    __shared__ int   sel[KTOP];

    for (int k = 0; k < KTOP; ++k) {
        float best = -3.402823466e+38f;
        int   bi   = 0x7fffffff;
        for (int i = tid; i < HM * HM; i += 256) {
            bool skip = false;
            for (int j = 0; j < k; ++j) if (sel[j] == i) skip = true;
            if (skip) continue;
            float v = h[i];
            if (v > best || (v == best && i < bi)) { best = v; bi = i; }
        }
        sval[tid] = best; sidx[tid] = bi;
        __syncthreads();
        for (int s = 128; s > 0; s >>= 1) {
            if (tid < s) {
                float ov = sval[tid + s]; int oi = sidx[tid + s];
                if (ov > sval[tid] || (ov == sval[tid] && oi < sidx[tid])) {
                    sval[tid] = ov; sidx[tid] = oi;
                }
            }
            __syncthreads();
        }
        if (tid == 0) sel[k] = sidx[0];
        __syncthreads();
    }

    if (tid < KTOP) {
        const int idx = sel[tid];
        const int yi = idx / HM, xi = idx % HM;
        const float yy = (float)yi, xx = (float)xi;
        const size_t plane = (size_t)HM * HM;
        const float* ob = off + (size_t)n * 2 * plane;
        const float* bb = bsz + (size_t)n * 4 * plane;
        const int p = yi * HM + xi;
        const float o0 = ob[p],             o1 = ob[plane + p];
        const float b0 = bb[p],             b1 = bb[plane + p];
        const float b2 = bb[2 * plane + p], b3 = bb[3 * plane + p];
        const float width  = b0 + b2;
        const float height = b1 + b3;
        const float cx = (xx + o0) * 4.0f;
        const float cy = (yy + o1) * 4.0f;
        const float X1 = (cx - width  * 0.5f) * 0.25f;
        const float Y1 = (cy - height * 0.5f) * 0.25f;
        const float X2 = (cx + width  * 0.5f) * 0.25f;
        const float Y2 = (cy + height * 0.5f) * 0.25f;
        const int rr = n * KTOP + tid;
        boxesH[rr * 5 + 0] = (float)n;
        boxesH[rr * 5 + 1] = X1;
        boxesH[rr * 5 + 2] = Y1;
        boxesH[rr * 5 + 3] = X2;
        boxesH[rr * 5 + 4] = Y2;
        boxesV[rr * 5 + 0] = (float)n;
        boxesV[rr * 5 + 1] = X1;
        boxesV[rr * 5 + 2] = 0.0f;
        boxesV[rr * 5 + 3] = X2;
        boxesV[rr * 5 + 4] = 64.0f;
    }
}

// ------- Stage 2: ROI-align (10x10 samples) + 2x2 mean + POOL(25x14) + embed -------
// Writes af in column-major [col = r*14 + t][c] layout, ready as WMMA B matrix.
__global__ void roialign_kernel(const float* __restrict__ featsH,
                                const float* __restrict__ featsV,
                                const float* __restrict__ boxesH,
                                const float* __restrict__ boxesV,
                                const float* __restrict__ embed_w,
                                const float* __restrict__ embed_b,
                                float* __restrict__ afH,
                                float* __restrict__ afV)
{
    const int r  = blockIdx.x >> 1;
    const int br = blockIdx.x & 1;
    const float* boxes = br ? boxesV : boxesH;
    const float* feats = br ? featsV : featsH;
    float*       af    = br ? afV    : afH;

    __shared__ int   syl[10], syh[10], sxl[10], sxh[10];
    __shared__ float sly[10], slx[10];
    __shared__ int   svy[10], svx[10];
    __shared__ float sbox[5];

    const int tid = threadIdx.x;
    if (tid < 5) sbox[tid] = boxes[r * 5 + tid];
    __syncthreads();

    const float x1 = sbox[1], y1 = sbox[2], x2 = sbox[3], y2 = sbox[4];
    const int bidx = (int)sbox[0];
    const float roiW = fmaxf(x2 - x1, 1.0f);
    const float roiH = fmaxf(y2 - y1, 1.0f);

    if (tid < 10) {
        const int j = tid;
        const float offj = (float)(j >> 1) + ((float)(j & 1) + 0.5f) * 0.5f;
        const float ys = y1 + offj * (roiH * 0.2f);
        const int vy = (ys > -1.0f) && (ys < (float)HM);
        const float y = fminf(fmaxf(ys, 0.0f), (float)(HM - 1));
        const int yl = (int)floorf(y);
        syl[j] = yl; syh[j] = min(yl + 1, HM - 1);
        sly[j] = y - (float)yl; svy[j] = vy;
    } else if (tid >= 16 && tid < 26) {
        const int j = tid - 16;
        const float offj = (float)(j >> 1) + ((float)(j & 1) + 0.5f) * 0.5f;
        const float xs = x1 + offj * (roiW * 0.2f);
        const int vx = (xs > -1.0f) && (xs < (float)HM);
        const float x = fminf(fmaxf(xs, 0.0f), (float)(HM - 1));
        const int xl = (int)floorf(x);
        sxl[j] = xl; sxh[j] = min(xl + 1, HM - 1);
        slx[j] = x - (float)xl; svx[j] = vx;
    }
    __syncthreads();

    const int c = tid; // 128 threads == 128 channels
    const float* base = feats + ((size_t)bidx * CCH + c) * (HM * HM);

    float pooled[25];
    #pragma unroll
    for (int p = 0; p < 25; ++p) pooled[p] = 0.0f;

    for (int jy = 0; jy < 10; ++jy) {
        const float* r0 = base + syl[jy] * HM;
        const float* r1 = base + syh[jy] * HM;
        const float ly = sly[jy];
        const int   vy = svy[jy];
        const int   py = (jy >> 1) * 5;
        for (int jx = 0; jx < 10; ++jx) {
            const int xl = sxl[jx], xh = sxh[jx];
            const float v00 = r0[xl], v01 = r0[xh];
            const float v10 = r1[xl], v11 = r1[xh];
            const float lx = slx[jx];
            float val = v00 * (1.0f - ly) * (1.0f - lx)
                      + v01 * (1.0f - ly) * lx
                      + v10 * ly * (1.0f - lx)
                      + v11 * ly * lx;
            val = (vy & svx[jx]) ? val : 0.0f;
            pooled[py + (jx >> 1)] += 0.25f * val;
        }
    }

    // positional embedding: emb[c] = b[c] + sum_d (box[d]/64) * W_e[c][d]
    float emb = embed_b[c];
    #pragma unroll
    for (int d = 0; d < 4; ++d)
        emb += (sbox[1 + d] * (1.0f / 64.0f)) * embed_w[c * 4 + d];

    // POOL: out[t] = mean(pooled[a..b)) with a = t*25/14, b = ceil((t+1)*25/14)
    #pragma unroll
    for (int t = 0; t < TT; ++t) {
        const int a = (t * 25) / 14;
        const int b = ((t + 1) * 25 + 13) / 14;
        float s = 0.0f;
        for (int p = a; p < b; ++p) s += pooled[p];
        s *= (1.0f / (float)(b - a));
        af[(size_t)(r * TT + t) * CCH + c] = s + emb; // coalesced across c
    }
}

// ------- Stage 3: fp32 WMMA GEMM (proj) for both branches + fused combine -------
// out[o, col] = sum_c W[o,c] * af[c, col] + bias[o];  e = (h*e^h + v*e^v)/(h+v)
__global__ void gemm_combine_kernel(const float* __restrict__ Wp,
                                    const float* __restrict__ bp,
                                    const float* __restrict__ afH,
                                    const float* __restrict__ afV,
                                    float* __restrict__ out,
                                    int cols)
{
    const int lane = threadIdx.x & 31;
    const int warp = threadIdx.x >> 5;
    const int tile = blockIdx.x * 4 + warp;
    const int oT = tile & 7;   // 128/16 = 8 row tiles
    const int cT = tile >> 3;

    const int lh = lane >> 4;  // half-wave: selects K pair {0,1} vs {2,3}
    const int lm = lane & 15;

    const int o_a = oT * 16 + lm;   // A row (M)
    const int col = cT * 16 + lm;   // B col (N)

    v8f accH = {0.f, 0.f, 0.f, 0.f, 0.f, 0.f, 0.f, 0.f};
    v8f accV = {0.f, 0.f, 0.f, 0.f, 0.f, 0.f, 0.f, 0.f};

    const float* wrow = Wp  + (size_t)o_a * CCH + lh * 2;
    const float* bhp  = afH + (size_t)col * CCH + lh * 2;
    const float* bvp  = afV + (size_t)col * CCH + lh * 2;

    for (int kt = 0; kt < CCH / 4; ++kt) {
        const v2f a  = *(const v2f*)(wrow + kt * 4);
        const v2f bh = *(const v2f*)(bhp  + kt * 4);
        const v2f bv = *(const v2f*)(bvp  + kt * 4);
        accH = __builtin_amdgcn_wmma_f32_16x16x4_f32(false, a, false, bh,
                                                     (short)0, accH, false, false);
        accV = __builtin_amdgcn_wmma_f32_16x16x4_f32(false, a, false, bv,
                                                     (short)0, accV, false, false);
    }

    #pragma unroll
    for (int i = 0; i < 8; ++i) {
        const int o = oT * 16 + i + 8 * lh;       // C/D layout: M = i + 8*(lane/16)
        const float bias = bp[o];
        const float h = accH[i] + bias;
        const float v = accV[i] + bias;
        const float e = (h * __expf(h) + v * __expf(v)) / (h + v);
        if (col < cols) {
            const int rr = col / TT;
            const int t  = col - rr * TT;
            out[(size_t)rr * (CCH * TT) + o * TT + t] = e;
        }
    }
}

extern "C" void kernel_launch(void* const* d_in, const int* in_sizes, int n_in,
                              void* d_out, int out_size, void* d_ws, size_t ws_size,
                              hipStream_t stream) {
    const float* heat = (const float*)d_in[0];  // (N,1,64,64)
    const float* off  = (const float*)d_in[1];  // (N,2,64,64)
    const float* bsz  = (const float*)d_in[2];  // (N,4,64,64)
    const float* fH   = (const float*)d_in[3];  // (N,128,64,64)
    const float* fV   = (const float*)d_in[4];  // (N,128,64,64)
    const float* ew   = (const float*)d_in[5];  // (128,4)
    const float* eb   = (const float*)d_in[6];  // (128,)
    const float* pw   = (const float*)d_in[7];  // (128,128)
    const float* pb   = (const float*)d_in[8];  // (128,)
    float* out = (float*)d_out;

    const int N = in_sizes[0] / (HM * HM);
    const int R = N * KTOP;
    const int cols = R * TT;
    const int colsPad = (cols + 15) & ~15;

    float* ws = (float*)d_ws;
    float* boxesH = ws;
    float* boxesV = boxesH + (size_t)R * 5;
    float* afH    = boxesV + (size_t)R * 5;
    float* afV    = afH + (size_t)colsPad * CCH;

    topk_boxes_kernel<<<N, 256, 0, stream>>>(heat, off, bsz, boxesH, boxesV);
    roialign_kernel<<<R * 2, CCH, 0, stream>>>(fH, fV, boxesH, boxesV, ew, eb, afH, afV);

    const int tiles = (CCH / 16) * (colsPad / 16); // 8 * 140 = 1120 for N=40
    gemm_combine_kernel<<<tiles / 4, 128, 0, stream>>>(pw, pb, afH, afV, out, cols);
}